// CRF_27865747816763
// MI455X (gfx1250) — compile-verified
//
#include <hip/hip_runtime.h>
#include <hip/hip_bf16.h>

// ---------------------------------------------------------------------------
// CRF: U = inputs @ W  (TDM -> LDS -> WMMA f32 GEMM), then per-batch
// Viterbi + logZ + loss.   B=256, N=128, M=768, K=40.
// U is (256, 130, 42) padded to stride 48 in workspace.
// ---------------------------------------------------------------------------

#define CRF_B      256
#define CRF_T      130               // N+2 rows of U per batch
#define CRF_M      768               // GEMM reduction dim
#define CRF_NC     42                // K+2 cols of U / W / E
#define CRF_NCP    48                // padded col stride in workspace
#define CRF_ROWS   (CRF_B * CRF_T)   // 33280
#define NSTATE     40
#define NT         128

#define KSTEPS     (CRF_M / 4)       // 192 WMMA k-steps
#define CHUNK_K    64                // k-columns staged per TDM op
#define NCHUNK     (CRF_M / CHUNK_K) // 12
#define WAVES_PB   4                 // waves per GEMM block

typedef __attribute__((ext_vector_type(2)))  float        v2f;
typedef __attribute__((ext_vector_type(8)))  float        v8f;
typedef __attribute__((ext_vector_type(4)))  unsigned int tdm_v4u;
typedef __attribute__((ext_vector_type(8)))  int          tdm_v8i;
typedef __attribute__((ext_vector_type(4)))  int          tdm_v4i;

#if defined(__has_builtin)
#if __has_builtin(__builtin_amdgcn_tensor_load_to_lds)
#define USE_TDM 1
#endif
#endif

// ---------------------------------------------------------------------------
// Kernel 0: pack W (768x42) into per-lane WMMA B-fragments, zero-padded to 48
// cols. Wp[(kstep*3 + tile)*32 + lane] = float2{ W[k][n], W[k+1][n] } where
// k = 4*kstep + 2*(lane>>4), n = 16*tile + (lane&15).  18432 float2 = 147 KB.
// ---------------------------------------------------------------------------
__global__ __launch_bounds__(256) void crf_pack_w(const float* __restrict__ W,
                                                  float2* __restrict__ Wp) {
  const int idx = blockIdx.x * 256 + threadIdx.x;
  if (idx >= KSTEPS * 3 * 32) return;
  const int lane  = idx & 31;
  const int rest  = idx >> 5;
  const int tile  = rest % 3;
  const int kstep = rest / 3;
  const int k = kstep * 4 + 2 * (lane >> 4);
  const int n = tile * 16 + (lane & 15);
  float2 v; v.x = 0.f; v.y = 0.f;
  if (n < CRF_NC) { v.x = W[k * CRF_NC + n]; v.y = W[(k + 1) * CRF_NC + n]; }
  Wp[idx] = v;
}

#if USE_TDM
// Build a 2-D Tensor DMA descriptor (ISA 8.3/8.4) and fire TENSOR_LOAD_TO_LDS:
// tile = 16 rows x 64 cols of f32, row stride 768, into LDS at lds_byte_off.
// This toolchain's builtin takes 6 args: (g0 v4u, g1 v8i, g2 v4i, g3 v4i,
// g4 v8i, cpol i32); groups 2/3/4 are unused for a 2-D tile -> zeros.
__device__ __forceinline__ void tdm_load_tile(const float* gsrc,
                                              unsigned lds_byte_off) {
  const unsigned long long ga = (unsigned long long)(uintptr_t)gsrc;
  tdm_v4u g0;
  g0[0] = 1u;                                            // count=1, user D#
  g0[1] = lds_byte_off;                                  // lds_addr [63:32]
  g0[2] = (unsigned)(ga & 0xFFFFFFFFu);                  // global_addr lo
  g0[3] = (unsigned)((ga >> 32) & 0x01FFFFFFu) | (2u << 30); // addr hi | type=2
  tdm_v8i g1;
  g1[0] = (2 << 16);                                     // data_size = 4 bytes
  g1[1] = (int)((CRF_M & 0xFFFF) << 16);                 // tensor_dim0 = 768 (lo16)
  g1[2] = (int)(((CRF_M >> 16) & 0xFFFF) | (16 << 16));  // dim0 hi | tensor_dim1=16
  g1[3] = (int)(CHUNK_K << 16);                          // dim1 hi=0 | tile_dim0=64
  g1[4] = 16;                                            // tile_dim1=16, tile_dim2=0
  g1[5] = CRF_M;                                         // tensor_dim0_stride = 768
  g1[6] = 0;
  g1[7] = 0;
  const tdm_v4i gz4 = {0, 0, 0, 0};
  const tdm_v8i gz8 = {0, 0, 0, 0, 0, 0, 0, 0};
  __builtin_amdgcn_tensor_load_to_lds(g0, g1, gz4, gz4, gz8, 0);
}
#endif

// ---------------------------------------------------------------------------
// Kernel 1: U = inputs @ W via V_WMMA_F32_16X16X4_F32.
// Grid: 520 blocks x 128 threads (4 waves). Each wave: one 16-row tile x
// all 3 col tiles. A staged via TDM into double-buffered LDS (32 KB/block).
// ---------------------------------------------------------------------------
__global__ __launch_bounds__(128) void crf_gemm_u(const float* __restrict__ A,
                                                  const float2* __restrict__ Wp,
                                                  float* __restrict__ U) {
  const int lane = threadIdx.x & 31;
  const int wave = threadIdx.x >> 5;
  const int row0 = (blockIdx.x * WAVES_PB + wave) * 16;
  const int half = lane >> 4;   // selects K-pair within a WMMA step
  const int mr   = lane & 15;   // A row / B col within tile

  v8f acc0 = {}, acc1 = {}, acc2 = {};

#if USE_TDM
  __shared__ float sA[WAVES_PB][2][16 * CHUNK_K];   // 32 KB, double buffered
  const float* __restrict__ abase = A + (size_t)row0 * CRF_M;
  const unsigned lds0 = (unsigned)(uintptr_t)&sA[wave][0][0];
  const unsigned lds1 = (unsigned)(uintptr_t)&sA[wave][1][0];

  tdm_load_tile(abase, lds0);                       // prime chunk 0
#pragma unroll
  for (int c = 0; c < NCHUNK; ++c) {
    if (c + 1 < NCHUNK) {                           // prefetch next chunk
      tdm_load_tile(abase + (c + 1) * CHUNK_K, (((c + 1) & 1) ? lds1 : lds0));
      __builtin_amdgcn_s_wait_tensorcnt(1);         // chunk c has landed
    } else {
      __builtin_amdgcn_s_wait_tensorcnt(0);
    }
    const float* __restrict__ lb = &sA[wave][c & 1][0] + mr * CHUNK_K + 2 * half;
    const float2* __restrict__ wpc = Wp + (size_t)(c * 16) * 96 + lane;
#pragma unroll
    for (int s = 0; s < 16; ++s) {
      const float2 af = *reinterpret_cast<const float2*>(lb + s * 4);
      v2f a; a.x = af.x; a.y = af.y;
      const float2 w0 = wpc[s * 96];
      const float2 w1 = wpc[s * 96 + 32];
      const float2 w2 = wpc[s * 96 + 64];
      v2f b0; b0.x = w0.x; b0.y = w0.y;
      v2f b1; b1.x = w1.x; b1.y = w1.y;
      v2f b2; b2.x = w2.x; b2.y = w2.y;
      acc0 = __builtin_amdgcn_wmma_f32_16x16x4_f32(false, a, false, b0, (short)0, acc0, false, false);
      acc1 = __builtin_amdgcn_wmma_f32_16x16x4_f32(false, a, false, b1, (short)0, acc1, false, false);
      acc2 = __builtin_amdgcn_wmma_f32_16x16x4_f32(false, a, false, b2, (short)0, acc2, false, false);
    }
  }
#else
  // Fallback: direct global A loads (branch-free W via pack).
  const float* __restrict__ arow = A + (size_t)(row0 + mr) * CRF_M + 2 * half;
#pragma unroll 4
  for (int ks = 0; ks < KSTEPS; ++ks) {
    const float2 af = *reinterpret_cast<const float2*>(arow + ks * 4);
    v2f a; a.x = af.x; a.y = af.y;
    const float2* __restrict__ wpc = Wp + (size_t)ks * 96 + lane;
    const float2 w0 = wpc[0];
    const float2 w1 = wpc[32];
    const float2 w2 = wpc[64];
    v2f b0; b0.x = w0.x; b0.y = w0.y;
    v2f b1; b1.x = w1.x; b1.y = w1.y;
    v2f b2; b2.x = w2.x; b2.y = w2.y;
    acc0 = __builtin_amdgcn_wmma_f32_16x16x4_f32(false, a, false, b0, (short)0, acc0, false, false);
    acc1 = __builtin_amdgcn_wmma_f32_16x16x4_f32(false, a, false, b1, (short)0, acc1, false, false);
    acc2 = __builtin_amdgcn_wmma_f32_16x16x4_f32(false, a, false, b2, (short)0, acc2, false, false);
  }
#endif

  // C/D layout: VGPR v, lanes 0-15 -> M=v,N=lane ; lanes 16-31 -> M=v+8,N=lane-16
#pragma unroll
  for (int v = 0; v < 8; ++v) {
    const size_t r = (size_t)(row0 + v + 8 * half) * CRF_NCP;
    U[r + mr]      = acc0[v];
    U[r + mr + 16] = acc1[v];
    U[r + mr + 32] = acc2[v];   // cols 42..47 land in workspace padding
  }
}

// ---------------------------------------------------------------------------
// Kernel 2: per-batch Viterbi (max-plus) + logZ (logsumexp) + F.
// Grid: 256 blocks x 64 threads. States handled by threads 0..39.
// ---------------------------------------------------------------------------
__global__ __launch_bounds__(64) void crf_batch(const float* __restrict__ Uws,
                                                const float* __restrict__ E,
                                                const int* __restrict__ Y,
                                                float* __restrict__ yhat,
                                                float* __restrict__ negll) {
  __shared__ float sPi[NT][NSTATE];         // forward max scores (history)
  __shared__ float sEin[NSTATE][NSTATE];    // E[1:-1,1:-1]
  __shared__ float sExpE[NSTATE][NSTATE];   // exp(Ein)
  __shared__ float sAlpha[2][NSTATE];       // logsumexp forward (double buffer)
  __shared__ float sP[NSTATE];              // exp(alpha - m)
  __shared__ float sRed[64];
  __shared__ float sLogZ;

  const int b   = blockIdx.x;
  const int tid = threadIdx.x;
  const float* __restrict__ Ub = Uws + (size_t)b * CRF_T * CRF_NCP;

  for (int idx = tid; idx < NSTATE * NSTATE; idx += 64) {
    const int i = idx / NSTATE, j = idx % NSTATE;
    const float e = E[(i + 1) * CRF_NC + (j + 1)];
    sEin[i][j]  = e;
    sExpE[i][j] = __expf(e);
  }
  if (tid < NSTATE) {
    const float u = Ub[1 * CRF_NCP + (tid + 1)];
    sPi[0][tid]    = u;
    sAlpha[0][tid] = u;
  }
  __syncthreads();

  int cur = 0;
  for (int t = 1; t < NT; ++t) {
    float m = 0.f;
    if (tid < NSTATE) {
      m = sAlpha[cur][0];
#pragma unroll
      for (int i = 1; i < NSTATE; ++i) m = fmaxf(m, sAlpha[cur][i]);
      sP[tid] = __expf(sAlpha[cur][tid] - m);
    }
    __syncthreads();
    if (tid < NSTATE) {
      const int j = tid;
      const float u = Ub[(size_t)(t + 1) * CRF_NCP + (j + 1)];
      float best = -__builtin_inff();
      float s = 0.f;
#pragma unroll 8
      for (int i = 0; i < NSTATE; ++i) {
        best = fmaxf(best, sPi[t - 1][i] + sEin[i][j]);
        s = fmaf(sP[i], sExpE[i][j], s);
      }
      sPi[t][j]          = best + u;
      sAlpha[cur ^ 1][j] = __logf(s) + m + u;
    }
    __syncthreads();
    cur ^= 1;
  }

  // logZ = logsumexp(alpha_final)
  if (tid == 0) {
    float m2 = sAlpha[cur][0];
    for (int i = 1; i < NSTATE; ++i) m2 = fmaxf(m2, sAlpha[cur][i]);
    float s2 = 0.f;
    for (int i = 0; i < NSTATE; ++i) s2 += __expf(sAlpha[cur][i] - m2);
    sLogZ = __logf(s2) + m2;
  }

  // Backtrace (thread 0): first-max tie-break matches jnp.argmax.
  if (tid == 0) {
    int bestj = 0; float bv = sPi[NT - 1][0];
    for (int jj = 1; jj < NSTATE; ++jj) {
      const float v = sPi[NT - 1][jj];
      if (v > bv) { bv = v; bestj = jj; }
    }
    int last = bestj + 1;
    float* __restrict__ yb = yhat + (size_t)b * CRF_T;
    yb[0]      = 0.f;
    yb[NT + 1] = (float)(NSTATE + 1);  // 41
    yb[NT]     = (float)last;
    for (int t = NT - 1; t >= 1; --t) {
      int bi = 0;
      float sv = sPi[t - 1][0] + sEin[0][last - 1];
      for (int i = 1; i < NSTATE; ++i) {
        const float v = sPi[t - 1][i] + sEin[i][last - 1];
        if (v > sv) { sv = v; bi = i; }
      }
      last = bi + 1;
      yb[t] = (float)last;
    }
  }

  // F = sum_t U[b,t,Y[t]] (t=1..128) + sum_t E[Y[t],Y[t+1]] (t=1..127)
  const int* __restrict__ Yb = Y + (size_t)b * CRF_T;
  float f = 0.f;
  for (int t = 1 + tid; t <= NT; t += 64)
    f += Ub[(size_t)t * CRF_NCP + Yb[t]];
  for (int t = 1 + tid; t <= NT - 1; t += 64)
    f += E[Yb[t] * CRF_NC + Yb[t + 1]];
  sRed[tid] = f;
  __syncthreads();
#pragma unroll
  for (int off = 32; off > 0; off >>= 1) {
    if (tid < off) sRed[tid] += sRed[tid + off];
    __syncthreads();
  }
  if (tid == 0) negll[b] = -(sRed[0] - sLogZ);
}

// ---------------------------------------------------------------------------
// Kernel 3: deterministic tree reduction of 256 per-batch values -> mean loss.
// ---------------------------------------------------------------------------
__global__ __launch_bounds__(256) void crf_loss(const float* __restrict__ negll,
                                                float* __restrict__ out) {
  __shared__ float s[256];
  s[threadIdx.x] = negll[threadIdx.x];
  __syncthreads();
#pragma unroll
  for (int off = 128; off > 0; off >>= 1) {
    if (threadIdx.x < off) s[threadIdx.x] += s[threadIdx.x + off];
    __syncthreads();
  }
  if (threadIdx.x == 0) out[0] = s[0] * (1.0f / (float)CRF_B);
}

// ---------------------------------------------------------------------------
extern "C" void kernel_launch(void* const* d_in, const int* in_sizes, int n_in,
                              void* d_out, int out_size, void* d_ws, size_t ws_size,
                              hipStream_t stream) {
  (void)in_sizes; (void)n_in; (void)out_size; (void)ws_size;
  const float* inp = (const float*)d_in[0];   // (256,130,768) f32
  const int*   Y   = (const int*)d_in[1];     // (256,130)     i32
  const float* W   = (const float*)d_in[2];   // (768,42)      f32
  const float* E   = (const float*)d_in[3];   // (42,42)       f32
  float* out = (float*)d_out;                 // [33280 Y_hat | 1 loss]

  float*  Uws   = (float*)d_ws;                          // (33280,48) f32
  float*  negll = Uws + (size_t)CRF_ROWS * CRF_NCP;      // 256 f32
  float2* Wpack = (float2*)(negll + 256);                // 18432 float2

  crf_pack_w<<<(KSTEPS * 3 * 32 + 255) / 256, 256, 0, stream>>>(W, Wpack);
  crf_gemm_u<<<CRF_ROWS / (16 * WAVES_PB), 128, 0, stream>>>(inp, Wpack, Uws);
  crf_batch<<<CRF_B, 64, 0, stream>>>(Uws, E, Y, out, negll);
  crf_loss<<<1, 256, 0, stream>>>(negll, out + CRF_ROWS);
}